// QuantumLLPModel_90640989815388
// MI455X (gfx1250) — compile-verified
//
#include <hip/hip_runtime.h>
#include <hip/hip_bf16.h>

typedef __attribute__((ext_vector_type(2))) float v2f;
typedef __attribute__((ext_vector_type(8))) float v8f;

#define PI_F 3.14159265358979323846f

// q-value for column j (j = 8*b3 + 4*b2 + 2*b1 + b0 selecting qubits 10..13),
// multiplied in the same association order as the reference kron loop.
__device__ __forceinline__ float qsel(int j, float prefix,
                                      float pa0, float pa1,
                                      float pb0, float pb1,
                                      float pc0, float pc1,
                                      float pd0, float pd1) {
  float q = prefix;
  q *= ((j >> 3) & 1) ? pa1 : pa0;   // qubit 10
  q *= ((j >> 2) & 1) ? pb1 : pb0;   // qubit 11
  q *= ((j >> 1) & 1) ? pc1 : pc0;   // qubit 12
  q *= ((j >> 0) & 1) ? pd1 : pd0;   // qubit 13
  return q;
}

// One wave32 handles 16 batch rows. Lanes l and l+16 both own row (l&15):
// lane<16 supplies K-slots {0,1} (cols 4c+0,4c+1), lane>=16 supplies K-slots
// {2,3} (cols 4c+2,4c+3) for the three K=4 WMMA chunks c=0,1,2 (cols 10,11 pad=0).
// B = all-ones => accumulated D has rowsum(q) in every column.
__global__ __launch_bounds__(256) void qllp_wmma_kernel(
    const float* __restrict__ x,    // [B,14]
    const float* __restrict__ prm,  // [14]
    float* __restrict__ out,        // [B,10]
    int B) {
  const int tid  = threadIdx.x;
  const int lane = tid & 31;
  const int wave = tid >> 5;
  const int r    = lane & 15;                       // row-in-wave
  const int row  = blockIdx.x * 128 + wave * 16 + r;
  const bool valid = row < B;
  const int rowc = valid ? row : (B - 1);           // clamp loads, keep EXEC full

  // Per-qubit probabilities: p0 = cos^2(a/2) = (1+cos a)/2, p1 = (1-cos a)/2.
  const float* xr = x + (long)rowc * 14;
  float p0[14], p1[14];
#pragma unroll
  for (int i = 0; i < 14; ++i) {
    float ang = PI_F * xr[i] + prm[i];              // prm[] uniform -> s_loads
    float c = __cosf(ang);
    p0[i] = 0.5f + 0.5f * c;
    p1[i] = 0.5f - 0.5f * c;
  }

  float prefix = p0[0];
#pragma unroll
  for (int i = 1; i < 10; ++i) prefix *= p0[i];
  if (!valid) prefix = 0.0f;                        // padded rows contribute 0

  const int off = (lane & 16) ? 2 : 0;              // K-slot column offset
  const float pa0 = p0[10], pa1 = p1[10];
  const float pb0 = p0[11], pb1 = p1[11];
  const float pc0 = p0[12], pc1 = p1[12];
  const float pd0 = p0[13], pd1 = p1[13];

  v2f a0, a1, a2;
  a0.x = qsel(off + 0, prefix, pa0, pa1, pb0, pb1, pc0, pc1, pd0, pd1);
  a0.y = qsel(off + 1, prefix, pa0, pa1, pb0, pb1, pc0, pc1, pd0, pd1);
  a1.x = qsel(4 + off + 0, prefix, pa0, pa1, pb0, pb1, pc0, pc1, pd0, pd1);
  a1.y = qsel(4 + off + 1, prefix, pa0, pa1, pb0, pb1, pc0, pc1, pd0, pd1);
  const int j2 = 8 + off;                           // 8 (lower lanes) or 10 (pad)
  a2.x = (j2 + 0 < 10) ? qsel(j2 + 0, prefix, pa0, pa1, pb0, pb1, pc0, pc1, pd0, pd1) : 0.0f;
  a2.y = (j2 + 1 < 10) ? qsel(j2 + 1, prefix, pa0, pa1, pb0, pb1, pc0, pc1, pd0, pd1) : 0.0f;

  // Rowsum via matrix pipe: D += A(16x4) * ones(4x16), chained over 3 K-chunks.
  v2f bones; bones.x = 1.0f; bones.y = 1.0f;
  v8f acc = {};
  acc = __builtin_amdgcn_wmma_f32_16x16x4_f32(false, a0, false, bones, (short)0, acc, false, false);
  acc = __builtin_amdgcn_wmma_f32_16x16x4_f32(false, a1, false, bones, (short)0, acc, false, false);
  acc = __builtin_amdgcn_wmma_f32_16x16x4_f32(false, a2, false, bones, (short)0, acc, false, false);

  // D layout: lane n<16 holds rowsum(m) in acc[m] (m=0..7); lanes>=16 hold
  // rowsum(m+8) in acc[m]. Extract own acc[lane&7], then ds_bpermute from the
  // lane/half that owns rowsum(lane&15).
  const int r3 = lane & 7;
  float s0 = (r3 & 1) ? acc[1] : acc[0];
  float s1 = (r3 & 1) ? acc[3] : acc[2];
  float s2 = (r3 & 1) ? acc[5] : acc[4];
  float t0 = (r3 & 2) ? s1 : s0;
  float s3 = (r3 & 1) ? acc[7] : acc[6];
  float t1 = (r3 & 2) ? s3 : s2;
  float e  = (r3 & 4) ? t1 : t0;                    // = rowsum[(lane&7) | ((lane&16)>>1)]
  const int src = (lane & 7) | ((lane & 8) << 1);   // lane holding rowsum[lane&15]
  const int eb = __builtin_amdgcn_ds_bpermute(src << 2, __float_as_int(e));
  const float rs = __int_as_float(eb);

  if (valid) {
    const float inv = 1.0f / rs;                    // one precise div chain per lane
    float* orow = out + (long)row * 10;
    float2 v;
    v.x = a0.x * inv; v.y = a0.y * inv;
    *reinterpret_cast<float2*>(orow + off) = v;         // cols {0,1} or {2,3}
    v.x = a1.x * inv; v.y = a1.y * inv;
    *reinterpret_cast<float2*>(orow + 4 + off) = v;     // cols {4,5} or {6,7}
    if (off == 0) {
      v.x = a2.x * inv; v.y = a2.y * inv;
      *reinterpret_cast<float2*>(orow + 8) = v;         // cols {8,9}
    }
  }
}

extern "C" void kernel_launch(void* const* d_in, const int* in_sizes, int n_in,
                              void* d_out, int out_size, void* d_ws, size_t ws_size,
                              hipStream_t stream) {
  const float* x   = (const float*)d_in[0];   // [B,14] float32
  const float* prm = (const float*)d_in[1];   // [14]   float32
  float* out = (float*)d_out;                 // [B,10] float32
  const int B = in_sizes[0] / 14;             // 8192
  const int rowsPerBlock = 128;               // 8 waves * 16 rows
  const int grid = (B + rowsPerBlock - 1) / rowsPerBlock;
  qllp_wmma_kernel<<<grid, 256, 0, stream>>>(x, prm, out, B);
}